// BarycentricInterpolate_3650722201690
// MI455X (gfx1250) — compile-verified
//
#include <hip/hip_runtime.h>
#include <math.h>

typedef __attribute__((ext_vector_type(2))) float v2f;
typedef __attribute__((ext_vector_type(8))) float v8f;

#define NODES   64
#define KSTEPS  16          // 64 nodes / K=4 per WMMA
#define QPT     16          // queries per WMMA tile (N dimension)
#define TPW     4           // tiles per wave, all in flight simultaneously
#define WAVES   8           // waves per block (256 threads, wave32)
#define FLT_BIG 3.402823466e+38f

// D(16x16) = A(16x4) x B(4x16) + C, all fp32 (V_WMMA_F32_16X16X4_F32).
// A: row0 = fi*wi, row1 = wi (constant across all tiles).
// B: c = 1/(x - xi), queries along N.
// D row0 = numer, row1 = denom -> lanes 0..15, acc[0]/acc[1].
__global__ __launch_bounds__(256)
void bary_wmma_kernel(const float* __restrict__ x,
                      const float* __restrict__ xi,
                      const float* __restrict__ fi,
                      const float* __restrict__ wi,
                      float* __restrict__ out,
                      int m_total)
{
    const int lane = threadIdx.x & 31;
    const int wave = threadIdx.x >> 5;
    const int m16  = lane & 15;
    const int hi2  = (lane >> 4) << 1;            // low half supplies K=0,1; high half K=2,3

    // ---- tile-invariant per-lane node data (kept in VGPRs for the whole wave) ----
    float xiv0[KSTEPS], xiv1[KSTEPS], a0[KSTEPS], a1[KSTEPS];
#pragma unroll
    for (int kb = 0; kb < KSTEPS; ++kb) {
        const int kv = kb * 4 + hi2;              // even -> 8B-aligned pair loads
        const float2 xp = *(const float2*)(xi + kv);
        const float2 wp = *(const float2*)(wi + kv);
        const float2 fp = *(const float2*)(fi + kv);
        xiv0[kb] = xp.x;  xiv1[kb] = xp.y;
        // A-matrix lane layout: lane m16 holds row M=m16; VGPR0=K even, VGPR1=K odd
        a0[kb] = (m16 == 0) ? fp.x * wp.x : ((m16 == 1) ? wp.x : 0.0f);
        a1[kb] = (m16 == 0) ? fp.y * wp.y : ((m16 == 1) ? wp.y : 0.0f);
    }

    const int waveId = blockIdx.x * WAVES + wave;

    // ---- 4 tiles in flight: 8 independent rcp chains + 4 WMMA acc chains per K-step ----
    int   qb[TPW];
    float xq[TPW];
    v8f   acc[TPW];
#pragma unroll
    for (int t = 0; t < TPW; ++t) {
        qb[t] = (waveId * TPW + t) * QPT;
        if (qb[t] > m_total - QPT) qb[t] = m_total - QPT;  // exact fit normally; benign clamp
        xq[t] = x[qb[t] + m16];
        acc[t] = (v8f){};
    }

#pragma unroll
    for (int kb = 0; kb < KSTEPS; ++kb) {
        v2f av; av.x = a0[kb]; av.y = a1[kb];
        float d0[TPW], d1[TPW], r0[TPW], r1[TPW];
#pragma unroll
        for (int t = 0; t < TPW; ++t) {
            d0[t] = xq[t] - xiv0[kb];
            d1[t] = xq[t] - xiv1[kb];
            r0[t] = __builtin_amdgcn_rcpf(d0[t]);
            r1[t] = __builtin_amdgcn_rcpf(d1[t]);
        }
#pragma unroll
        for (int t = 0; t < TPW; ++t) {
            // one Newton step -> ~IEEE-quality reciprocal; exact hit (d==0) -> NaN,
            // caught by the finite check after the loop.
            r0[t] = fmaf(fmaf(-d0[t], r0[t], 1.0f), r0[t], r0[t]);
            r1[t] = fmaf(fmaf(-d1[t], r1[t], 1.0f), r1[t], r1[t]);
        }
#pragma unroll
        for (int t = 0; t < TPW; ++t) {
            v2f bv; bv.x = r0[t]; bv.y = r1[t];
            // (neg_a, A, neg_b, B, c_mod, C, reuse_a, reuse_b)
            acc[t] = __builtin_amdgcn_wmma_f32_16x16x4_f32(false, av, false, bv,
                                                           (short)0, acc[t], false, false);
        }
    }

    if (lane < 16) {                              // D rows 0/1 live in lanes 0..15
#pragma unroll
        for (int t = 0; t < TPW; ++t) {
            float nmr = acc[t][0], dnm = acc[t][1];
            float res = nmr / dnm;
            // rare exact-hit fallback (non-finite accumulators); matches argmax(first-hit)
            if (!(fabsf(nmr) <= FLT_BIG && fabsf(dnm) <= FLT_BIG)) {
                for (int j = 0; j < NODES; ++j)
                    if (xq[t] - xi[j] == 0.0f) { res = fi[j]; break; }
            }
            out[qb[t] + m16] = res;
        }
    }
}

extern "C" void kernel_launch(void* const* d_in, const int* in_sizes, int n_in,
                              void* d_out, int out_size, void* d_ws, size_t ws_size,
                              hipStream_t stream) {
    const float* x  = (const float*)d_in[0];
    const float* xi = (const float*)d_in[1];
    const float* fi = (const float*)d_in[2];
    const float* wi = (const float*)d_in[3];
    float* out = (float*)d_out;
    const int m = in_sizes[0];                    // 2,097,152
    const int qPerBlock = QPT * TPW * WAVES;      // 512 queries/block
    const int blocks = (m + qPerBlock - 1) / qPerBlock;   // 4096
    hipLaunchKernelGGL(bary_wmma_kernel, dim3(blocks), dim3(256), 0, stream,
                       x, xi, fi, wi, out, m);
}